// Edge_BCELoss_24489903522032
// MI455X (gfx1250) — compile-verified
//
#include <hip/hip_runtime.h>
#include <hip/hip_bf16.h>
#include <stdint.h>

#define NIMG 16
#define H 384
#define W 384
#define NHW (NIMG * H * W)        // 2359296
#define BIGV 10000.0f

#define COLS 32                    // columns per tile
#define RPW  48                    // output rows per workgroup
#define CT   (W / COLS)            // 12 column tiles
#define RB   (H / RPW)             // 8 row blocks
#define K2_BLOCKS (NIMG * CT * RB) // 1536

#define K3_BLOCKS 1152
#define K3_ITERS  (NHW / (K3_BLOCKS * 256)) // 8 exactly

typedef __attribute__((ext_vector_type(2))) float v2f;
typedef __attribute__((ext_vector_type(8))) float v8f;
typedef int v4i __attribute__((vector_size(16)));
typedef __attribute__((address_space(1))) v4i* g_v4i_ptr;   // global (device) b128 pointer
typedef __attribute__((address_space(3))) v4i* l_v4i_ptr;   // LDS b128 pointer

// ---------- CDNA5 async global->LDS copy (b128) ----------
__device__ __forceinline__ void async_copy_b128(const float* gsrc, float* ldst) {
#if __has_builtin(__builtin_amdgcn_global_load_async_to_lds_b128)
  __builtin_amdgcn_global_load_async_to_lds_b128(
      (g_v4i_ptr)(v4i*)gsrc,
      (l_v4i_ptr)(v4i*)ldst,
      0, 0);
#else
  unsigned loff = (unsigned)(uintptr_t)(void*)ldst;   // low 32 bits of flat LDS addr = LDS offset
  asm volatile("global_load_async_to_lds_b128 %0, %1, off"
               :: "v"(loff), "v"(gsrc) : "memory");
#endif
}

__device__ __forceinline__ void wait_async0() {
#if __has_builtin(__builtin_amdgcn_s_wait_asynccnt)
  __builtin_amdgcn_s_wait_asynccnt(0);
#else
  asm volatile("s_wait_asynccnt 0" ::: "memory");
#endif
}

// ---------- Kernel 1: per-row horizontal distance -> g2; init min/max ----------
__global__ void __launch_bounds__(256)
k_rowdist(const float* __restrict__ edge, float* __restrict__ g2,
          unsigned* __restrict__ minU, unsigned* __restrict__ maxU) {
  int r = blockIdx.x * blockDim.x + threadIdx.x;
  if (r < NIMG) { minU[r] = 0x7f800000u; maxU[r] = 0u; }  // +inf / 0.0
  if (r >= NIMG * H) return;
  const float* e = edge + (size_t)r * W;
  float* g = g2 + (size_t)r * W;
  // backward pass: distance to nearest zero at/right (BIG-ish if none)
  float nxt = BIGV;  // matches reference: no zero -> nxt = +BIG
  for (int x = W - 1; x >= 0; --x) {
    if (e[x] == 0.0f) nxt = (float)x;
    g[x] = nxt - (float)x;
  }
  // forward pass: combine with distance to nearest zero at/left, clamp, square
  float last = -BIGV;
  for (int x = 0; x < W; ++x) {
    if (e[x] == 0.0f) last = (float)x;
    float gl = (float)x - last;
    float gg = fminf(fminf(g[x], gl), BIGV);
    g[x] = gg * gg;
  }
}

// ---------- Kernel 2: column min-plus EDT, sqrt, per-image min/max ----------
__global__ void __launch_bounds__(256)
k_edt(const float* __restrict__ g2, float* __restrict__ dist,
      unsigned* __restrict__ minU, unsigned* __restrict__ maxU) {
  __shared__ float tile[H * COLS];   // 48 KB: full 384-row strip of 32 columns
  __shared__ float sred[256];

  int wg  = blockIdx.x;
  int n   = wg / (CT * RB);
  int rm  = wg % (CT * RB);
  int ct  = rm / RB;
  int rb  = rm % RB;
  int x0  = ct * COLS;
  int i0  = rb * RPW;
  int tid = threadIdx.x;

  // async-load the g2 strip: 384*32 floats = 3072 x b128, 12 per thread
  const float* gbase = g2 + (size_t)n * H * W;
  #pragma unroll
  for (int k = 0; k < (H * COLS / 4) / 256; ++k) {
    int c  = k * 256 + tid;       // chunk id, 8 chunks per row
    int j  = c >> 3;
    int cx = (c & 7) * 4;
    async_copy_b128(gbase + (size_t)j * W + x0 + cx, &tile[j * COLS + cx]);
  }
  wait_async0();
  __syncthreads();

  float lmin = 3.0e38f, lmax = 0.0f;
  #pragma unroll
  for (int k = 0; k < (RPW * COLS) / 256; ++k) {   // 6 outputs per thread
    int o  = k * 256 + tid;
    int il = o >> 5;
    int xl = o & 31;
    int i  = i0 + il;
    float m = 3.0e38f;
    float d = (float)i;                 // (i - j), decremented each step; exact in f32
    #pragma unroll 4
    for (int j = 0; j < H; ++j) {
      float t = tile[j * COLS + xl] + d * d;
      m = fminf(m, t);
      d -= 1.0f;
    }
    float dv = sqrtf(m);
    dist[((size_t)n * H + i) * W + x0 + xl] = dv;
    lmin = fminf(lmin, dv);
    lmax = fmaxf(lmax, dv);
  }

  // block min reduction
  sred[tid] = lmin; __syncthreads();
  for (int s = 128; s > 0; s >>= 1) { if (tid < s) sred[tid] = fminf(sred[tid], sred[tid + s]); __syncthreads(); }
  float bmin = sred[0]; __syncthreads();
  // block max reduction
  sred[tid] = lmax; __syncthreads();
  for (int s = 128; s > 0; s >>= 1) { if (tid < s) sred[tid] = fmaxf(sred[tid], sred[tid + s]); __syncthreads(); }
  if (tid == 0) {
    // dist >= 0, so IEEE order == unsigned bit-pattern order (deterministic atomics)
    atomicMin(&minU[n], __float_as_uint(bmin));
    atomicMax(&maxU[n], __float_as_uint(sred[0]));
  }
}

// ---------- Kernel 3: weighted BCE partial sums ----------
__global__ void __launch_bounds__(256)
k_loss(const float* __restrict__ pred, const float* __restrict__ pred_edge,
       const float* __restrict__ label, const float* __restrict__ edge,
       const float* __restrict__ dist, const unsigned* __restrict__ minU,
       const unsigned* __restrict__ maxU, float* __restrict__ partials) {
  __shared__ float sred[256];
  int tid = threadIdx.x;
  size_t base = (size_t)blockIdx.x * 256 + tid;
  float acc = 0.0f;
  #pragma unroll
  for (int k = 0; k < K3_ITERS; ++k) {
    size_t idx = base + (size_t)k * (K3_BLOCKS * 256);
    int n = (int)(idx / (size_t)(H * W));
    float p  = pred[idx];
    float l  = label[idx];
    float pe = pred_edge[idx];
    float ed = edge[idx];
    float dv = dist[idx];
    float mn = __uint_as_float(minU[n]);
    float mx = __uint_as_float(maxU[n]);
    float w  = (dv - mn) / (mx - mn + 1e-8f);
    float lt = -(l * logf(p) + (1.0f - l) * logf(1.0f - p)) * (1.0f + w);
    float et = -(ed * logf(pe) + (1.0f - ed) * logf(1.0f - pe));
    acc += lt + et;
  }
  sred[tid] = acc; __syncthreads();
  for (int s = 128; s > 0; s >>= 1) { if (tid < s) sred[tid] += sred[tid + s]; __syncthreads(); }
  if (tid == 0) partials[blockIdx.x] = sred[0];
}

// ---------- Kernel 4: deterministic final reduction via WMMA ----------
// A[m][0] = s(lane m), A[m][2] = s(lane m+16), B = ones 4x16:
// D[m][n] = s(m) + s(m+16). Per-lane sum of the 8 D regs gives the half-sum of
// rowsums; one shfl_xor(16) completes the full 32-lane sum.
__global__ void __launch_bounds__(32)
k_final(const float* __restrict__ partials, float* __restrict__ out) {
  int lane = threadIdx.x;
  float s = 0.0f;
  #pragma unroll
  for (int k = 0; k < K3_BLOCKS / 32; ++k) s += partials[lane + k * 32];

  v2f a; a[0] = s;    a[1] = 0.0f;
  v2f b; b[0] = 1.0f; b[1] = 1.0f;
  v8f c = {0.0f, 0.0f, 0.0f, 0.0f, 0.0f, 0.0f, 0.0f, 0.0f};
  c = __builtin_amdgcn_wmma_f32_16x16x4_f32(false, a, false, b, (short)0, c, false, false);
  float t = c[0] + c[1] + c[2] + c[3] + c[4] + c[5] + c[6] + c[7];
  t += __shfl_xor(t, 16, 32);
  if (lane == 0) out[0] = t * (1.0f / (float)NHW);
}

extern "C" void kernel_launch(void* const* d_in, const int* in_sizes, int n_in,
                              void* d_out, int out_size, void* d_ws, size_t ws_size,
                              hipStream_t stream) {
  const float* pred      = (const float*)d_in[0];
  const float* pred_edge = (const float*)d_in[1];
  const float* label     = (const float*)d_in[2];
  const float* edge      = (const float*)d_in[3];

  float*    g2       = (float*)d_ws;
  float*    dist     = g2 + NHW;
  unsigned* minU     = (unsigned*)(dist + NHW);
  unsigned* maxU     = minU + NIMG;
  float*    partials = (float*)(maxU + NIMG);

  k_rowdist<<<(NIMG * H) / 256, 256, 0, stream>>>(edge, g2, minU, maxU);
  k_edt    <<<K2_BLOCKS,        256, 0, stream>>>(g2, dist, minU, maxU);
  k_loss   <<<K3_BLOCKS,        256, 0, stream>>>(pred, pred_edge, label, edge, dist,
                                                  minU, maxU, partials);
  k_final  <<<1,                 32, 0, stream>>>(partials, (float*)d_out);
}